// GATModule_18940805775664
// MI455X (gfx1250) — compile-verified
//
#include <hip/hip_runtime.h>
#include <math.h>

#define NNODES 30000
#define NEDGES 480000
#define ETOT   (NNODES + NEDGES)
#define DOUT   256
#define NEG_SLOPE 0.2f

typedef float v2f __attribute__((ext_vector_type(2)));
typedef float v8f __attribute__((ext_vector_type(8)));

// ---------------------------------------------------------------- utilities
__global__ void fill_kernel(float* __restrict__ p, long n, float v) {
  long i = (long)blockIdx.x * blockDim.x + threadIdx.x;
  if (i < n) p[i] = v;
}

__device__ __forceinline__ float leaky(float x) {
  return x > 0.f ? x : NEG_SLOPE * x;
}

__device__ __forceinline__ void edge_sd(const int* __restrict__ ei, int e,
                                        int& s, int& d) {
  if (e < NEDGES) { s = ei[e]; d = ei[NEDGES + e]; }
  else            { s = e - NEDGES; d = s; }          // appended self-loops
}

// float atomic max via signed/unsigned integer atomics (init must be -inf)
__device__ __forceinline__ void atomicMaxF(float* a, float v) {
  if (v >= 0.f) atomicMax((int*)a, __float_as_int(v));
  else          atomicMin((unsigned int*)a, __float_as_uint(v));
}

// ---------------------------------------------------------------- WMMA GEMM
// C[M,Nn] = A[M,K] @ B[K,Nn], f32, V_WMMA_F32_16X16X4_F32.
// Block: 256 threads = 8 waves; block tile 32x128; each wave owns two 16x16
// C tiles (cols bc and bc+64) sharing one A fragment.
// Main K-loop covers only FULL 16-chunks (no guards); tail chunk is peeled.
__global__ __launch_bounds__(256) void gemm_wmma_f32(
    const float* __restrict__ A, const float* __restrict__ B,
    float* __restrict__ C, int M, int K, int Nn) {
  __shared__ float As[32][17];    // 32 rows x 16 k (+1 pad)
  __shared__ float Bs[16][129];   // 16 k x 128 cols (+1 pad)

  const int tid   = threadIdx.x;
  const int lane  = tid & 31;
  const int wave  = tid >> 5;
  const int waveM = wave >> 2;        // 0..1
  const int waveN = wave & 3;         // 0..3
  const int half  = lane >> 4;        // K-half selector per ISA layout
  const int lr    = lane & 15;
  const int row0  = blockIdx.x * 32;
  const int col0  = blockIdx.y * 128;

  // staging indices: A by 256 threads x float2; B by 256 threads x 2*float4
  const int atr = tid >> 3;            // 0..31
  const int atk = (tid & 7) << 1;      // 0,2,..,14
  const int bkr = tid >> 4;            // 0..15
  const int bnc = (tid & 15) << 3;     // 0..120 step 8

  const int ar = (waveM << 4) + lr;    // A fragment row in LDS
  const int bc = (waveN << 4) + lr;    // B fragment col in LDS (first tile)

  v8f acc0 = {0.f, 0.f, 0.f, 0.f, 0.f, 0.f, 0.f, 0.f};
  v8f acc1 = {0.f, 0.f, 0.f, 0.f, 0.f, 0.f, 0.f, 0.f};

  // clamp A row (garbage rows only feed guarded C rows)
  int arow = row0 + atr; if (arow > M - 1) arow = M - 1;
  const float* aBase = A + (size_t)arow * K;
  const float* bBase = B + col0 + bnc + (size_t)bkr * Nn;

  const int Kfull = K & ~15;           // K rounded down to multiple of 16

  for (int k0 = 0; k0 < Kfull; k0 += 16) {
    {   // stage A 32x16 : unconditional b64 per thread
      const float2 v = *(const float2*)(aBase + k0 + atk);
      As[atr][atk]     = v.x;
      As[atr][atk + 1] = v.y;
    }
    {   // stage B 16x128 : unconditional 2 x b128 per thread
      const float* bp = bBase + (size_t)k0 * Nn;
      const float4 v0 = *(const float4*)bp;
      const float4 v1 = *(const float4*)(bp + 4);
      Bs[bkr][bnc]     = v0.x; Bs[bkr][bnc + 1] = v0.y;
      Bs[bkr][bnc + 2] = v0.z; Bs[bkr][bnc + 3] = v0.w;
      Bs[bkr][bnc + 4] = v1.x; Bs[bkr][bnc + 5] = v1.y;
      Bs[bkr][bnc + 6] = v1.z; Bs[bkr][bnc + 7] = v1.w;
    }
    __syncthreads();
#pragma unroll
    for (int s = 0; s < 4; ++s) {
      const int ka = (s << 2) + (half << 1); // lanes 0-15: K,K+1 ; 16-31: K+2,K+3
      v2f a, b0, b1;
      a.x  = As[ar][ka];          a.y  = As[ar][ka + 1];
      b0.x = Bs[ka][bc];          b0.y = Bs[ka + 1][bc];
      b1.x = Bs[ka][bc + 64];     b1.y = Bs[ka + 1][bc + 64];
      acc0 = __builtin_amdgcn_wmma_f32_16x16x4_f32(
          false, a, false, b0, (short)0, acc0, false, false);
      acc1 = __builtin_amdgcn_wmma_f32_16x16x4_f32(
          false, a, false, b1, (short)0, acc1, false, false);
    }
    __syncthreads();
  }

  if (Kfull < K) {                     // peeled tail chunk (guards live here only)
    {
      float v0 = 0.f, v1 = 0.f;
      if (Kfull + atk     < K) v0 = aBase[Kfull + atk];
      if (Kfull + atk + 1 < K) v1 = aBase[Kfull + atk + 1];
      As[atr][atk]     = v0;
      As[atr][atk + 1] = v1;
    }
    {
      float4 v0 = {0.f, 0.f, 0.f, 0.f}, v1 = {0.f, 0.f, 0.f, 0.f};
      if (Kfull + bkr < K) {
        const float* bp = bBase + (size_t)Kfull * Nn;
        v0 = *(const float4*)bp;
        v1 = *(const float4*)(bp + 4);
      }
      Bs[bkr][bnc]     = v0.x; Bs[bkr][bnc + 1] = v0.y;
      Bs[bkr][bnc + 2] = v0.z; Bs[bkr][bnc + 3] = v0.w;
      Bs[bkr][bnc + 4] = v1.x; Bs[bkr][bnc + 5] = v1.y;
      Bs[bkr][bnc + 6] = v1.z; Bs[bkr][bnc + 7] = v1.w;
    }
    __syncthreads();
#pragma unroll
    for (int s = 0; s < 4; ++s) {
      const int ka = (s << 2) + (half << 1);
      v2f a, b0, b1;
      a.x  = As[ar][ka];          a.y  = As[ar][ka + 1];
      b0.x = Bs[ka][bc];          b0.y = Bs[ka + 1][bc];
      b1.x = Bs[ka][bc + 64];     b1.y = Bs[ka + 1][bc + 64];
      acc0 = __builtin_amdgcn_wmma_f32_16x16x4_f32(
          false, a, false, b0, (short)0, acc0, false, false);
      acc1 = __builtin_amdgcn_wmma_f32_16x16x4_f32(
          false, a, false, b1, (short)0, acc1, false, false);
    }
    __syncthreads();
  }

  // C/D layout: VGPR r -> (M=r, lanes 0-15) / (M=r+8, lanes 16-31)
  const int rowb = row0 + (waveM << 4) + (half << 3);
  const int col  = col0 + bc;
#pragma unroll
  for (int r = 0; r < 8; ++r) {
    int row = rowb + r;
    if (row < M) {
      C[(size_t)row * Nn + col]      = acc0[r];
      C[(size_t)row * Nn + col + 64] = acc1[r];
    }
  }
}

// ------------------------------------------------------- attention scores
// One wave per node; each lane owns 8 contiguous channels of the 256-wide row.
// a_src/a_dst are flat [H*C]==[256]; reduction group g = C/8 lanes per head.
__global__ void scores_kernel(const float* __restrict__ h,
                              const float* __restrict__ asrc,
                              const float* __restrict__ adst,
                              float* __restrict__ s_src,
                              float* __restrict__ s_dst,
                              int Hh, int C) {
  int wave = blockIdx.x * (blockDim.x >> 5) + (threadIdx.x >> 5);
  if (wave >= NNODES) return;                 // wave-uniform
  int lane = threadIdx.x & 31;
  const float* row = h + (size_t)wave * DOUT + lane * 8;
  float ps = 0.f, pd = 0.f;
#pragma unroll
  for (int i = 0; i < 8; ++i) {
    float v = row[i];
    ps += v * asrc[lane * 8 + i];
    pd += v * adst[lane * 8 + i];
  }
  const int g = C >> 3;                        // lanes per head (8 or 32)
  for (int off = g >> 1; off > 0; off >>= 1) {
    ps += __shfl_xor(ps, off, 32);
    pd += __shfl_xor(pd, off, 32);
  }
  if ((lane & (g - 1)) == 0) {
    int head = lane / g;
    s_src[wave * Hh + head] = ps;
    s_dst[wave * Hh + head] = pd;
  }
}

// ------------------------------------------------------- edge softmax passes
__global__ void edge_max_kernel(const int* __restrict__ ei,
                                const float* __restrict__ ssrc,
                                const float* __restrict__ sdst,
                                float* __restrict__ m, int Hh) {
  long idx = (long)blockIdx.x * blockDim.x + threadIdx.x;
  if (idx >= (long)ETOT * Hh) return;
  int e = (int)(idx / Hh), hh = (int)(idx % Hh);
  int s, d; edge_sd(ei, e, s, d);
  float v = leaky(ssrc[s * Hh + hh] + sdst[d * Hh + hh]);
  atomicMaxF(&m[d * Hh + hh], v);
}

__global__ void edge_sum_kernel(const int* __restrict__ ei,
                                const float* __restrict__ ssrc,
                                const float* __restrict__ sdst,
                                const float* __restrict__ m,
                                float* __restrict__ den, int Hh) {
  long idx = (long)blockIdx.x * blockDim.x + threadIdx.x;
  if (idx >= (long)ETOT * Hh) return;
  int e = (int)(idx / Hh), hh = (int)(idx % Hh);
  int s, d; edge_sd(ei, e, s, d);
  float v = leaky(ssrc[s * Hh + hh] + sdst[d * Hh + hh]);
  atomicAdd(&den[d * Hh + hh], __expf(v - m[d * Hh + hh]));
}

// One wave per edge; 8 channels per lane; scatter-add into acc (L2 atomics).
__global__ void edge_agg_kernel(const int* __restrict__ ei,
                                const float* __restrict__ hbuf,
                                const float* __restrict__ ssrc,
                                const float* __restrict__ sdst,
                                const float* __restrict__ m,
                                const float* __restrict__ den,
                                float* __restrict__ acc, int Hh, int C) {
  int wave = blockIdx.x * (blockDim.x >> 5) + (threadIdx.x >> 5);
  if (wave >= ETOT) return;                    // wave-uniform
  int lane = threadIdx.x & 31;
  int s, d; edge_sd(ei, wave, s, d);
  int hh = (lane << 3) / C;                    // head owning these 8 channels
  float v = leaky(ssrc[s * Hh + hh] + sdst[d * Hh + hh]);
  float alpha = __expf(v - m[d * Hh + hh]) / den[d * Hh + hh];
  const float* hr = hbuf + (size_t)s * DOUT + lane * 8;
  float*       ac = acc  + (size_t)d * DOUT + lane * 8;
#pragma unroll
  for (int i = 0; i < 8; ++i) atomicAdd(&ac[i], alpha * hr[i]);
}

// ------------------------------------------------------- elementwise tails
__global__ void relu_bias_kernel(float* __restrict__ a,
                                 const float* __restrict__ b) {
  long i = (long)blockIdx.x * blockDim.x + threadIdx.x;
  if (i >= (long)NNODES * DOUT) return;
  float v = a[i] + b[i & 255];
  a[i] = v > 0.f ? v : 0.f;
}

__global__ void bias_out_kernel(const float* __restrict__ a,
                                const float* __restrict__ b,
                                float* __restrict__ o) {
  long i = (long)blockIdx.x * blockDim.x + threadIdx.x;
  if (i >= (long)NNODES * DOUT) return;
  o[i] = a[i] + b[i & 255];
}

// ---------------------------------------------------------------- driver
extern "C" void kernel_launch(void* const* d_in, const int* in_sizes, int n_in,
                              void* d_out, int out_size, void* d_ws, size_t ws_size,
                              hipStream_t stream) {
  const float* x   = (const float*)d_in[0];
  const int*   ei  = (const int*)  d_in[1];   // [2,E] flat: row0=src, row1=dst
  const float* W1  = (const float*)d_in[2];
  const float* as1 = (const float*)d_in[3];
  const float* ad1 = (const float*)d_in[4];
  const float* b1  = (const float*)d_in[5];
  const float* W2  = (const float*)d_in[6];
  const float* as2 = (const float*)d_in[7];
  const float* ad2 = (const float*)d_in[8];
  const float* b2  = (const float*)d_in[9];
  float* out = (float*)d_out;

  float* ws   = (float*)d_ws;
  float* bufA = ws;                                   // h1, later h2   [N,256]
  float* bufB = bufA + (size_t)NNODES * DOUT;         // acc1/h, acc2   [N,256]
  float* ss1  = bufB + (size_t)NNODES * DOUT;         // [N,4]
  float* sd1  = ss1 + (size_t)NNODES * 4;
  float* m1   = sd1 + (size_t)NNODES * 4;
  float* dn1  = m1  + (size_t)NNODES * 4;
  float* ss2  = dn1 + (size_t)NNODES * 4;             // [N]
  float* sd2  = ss2 + NNODES;
  float* m2   = sd2 + NNODES;
  float* dn2  = m2  + NNODES;

  const int TB = 256;
  const long nelem = (long)NNODES * DOUT;
  const int  nblk  = (int)((nelem + TB - 1) / TB);
  const int  nodeWaveBlk = (NNODES * 32 + TB - 1) / TB;
  const int  edgeWaveBlk = (ETOT   * 32 + TB - 1) / TB;
  dim3 gg((NNODES + 31) / 32, DOUT / 128);

  // ---- layer 1 ----
  fill_kernel<<<nblk, TB, 0, stream>>>(bufB, nelem, 0.f);
  fill_kernel<<<(NNODES * 4 + TB - 1) / TB, TB, 0, stream>>>(m1, (long)NNODES * 4, -INFINITY);
  fill_kernel<<<(NNODES * 4 + TB - 1) / TB, TB, 0, stream>>>(dn1, (long)NNODES * 4, 0.f);

  gemm_wmma_f32<<<gg, TB, 0, stream>>>(x, W1, bufA, NNODES, 1000, DOUT);
  scores_kernel<<<nodeWaveBlk, TB, 0, stream>>>(bufA, as1, ad1, ss1, sd1, 4, 64);

  long eh1 = (long)ETOT * 4;
  edge_max_kernel<<<(int)((eh1 + TB - 1) / TB), TB, 0, stream>>>(ei, ss1, sd1, m1, 4);
  edge_sum_kernel<<<(int)((eh1 + TB - 1) / TB), TB, 0, stream>>>(ei, ss1, sd1, m1, dn1, 4);
  edge_agg_kernel<<<edgeWaveBlk, TB, 0, stream>>>(ei, bufA, ss1, sd1, m1, dn1, bufB, 4, 64);

  relu_bias_kernel<<<nblk, TB, 0, stream>>>(bufB, b1);

  // ---- layer 2 ----
  gemm_wmma_f32<<<gg, TB, 0, stream>>>(bufB, W2, bufA, NNODES, DOUT, DOUT);
  scores_kernel<<<nodeWaveBlk, TB, 0, stream>>>(bufA, as2, ad2, ss2, sd2, 1, 256);

  fill_kernel<<<nblk, TB, 0, stream>>>(bufB, nelem, 0.f);     // acc2 (h consumed)
  fill_kernel<<<(NNODES + TB - 1) / TB, TB, 0, stream>>>(m2, NNODES, -INFINITY);
  fill_kernel<<<(NNODES + TB - 1) / TB, TB, 0, stream>>>(dn2, NNODES, 0.f);

  edge_max_kernel<<<(ETOT + TB - 1) / TB, TB, 0, stream>>>(ei, ss2, sd2, m2, 1);
  edge_sum_kernel<<<(ETOT + TB - 1) / TB, TB, 0, stream>>>(ei, ss2, sd2, m2, dn2, 1);
  edge_agg_kernel<<<edgeWaveBlk, TB, 0, stream>>>(ei, bufA, ss2, sd2, m2, dn2, bufB, 1, 256);

  bias_out_kernel<<<nblk, TB, 0, stream>>>(bufB, b2, out);
}